// LightGCN_68771016344237
// MI455X (gfx1250) — compile-verified
//
#include <hip/hip_runtime.h>

typedef float v2f __attribute__((ext_vector_type(2)));
typedef float v8f __attribute__((ext_vector_type(8)));

#define EMB_DIM 128

// ---------------- K1: deg[i] = 1 (self loop) ----------------
__global__ void k_deg_init(unsigned* __restrict__ deg, int N) {
    int i = blockIdx.x * blockDim.x + threadIdx.x;
    if (i < N) deg[i] = 1u;
}

// ---------------- K2: deg[row[e]] += 1 ----------------
__global__ void k_deg_count(const int* __restrict__ row, unsigned* __restrict__ deg, int E) {
    int e = blockIdx.x * blockDim.x + threadIdx.x;
    if (e < E) atomicAdd(&deg[row[e]], 1u);
}

// ---------------- K3: dinv = rsqrt(deg) ----------------
__global__ void k_dinv(const unsigned* __restrict__ deg, float* __restrict__ dinv, int N) {
    int i = blockIdx.x * blockDim.x + threadIdx.x;
    if (i < N) dinv[i] = rsqrtf((float)deg[i]);
}

// ---------------- K4: out[i] = dinv[i]^2 * emb[i]  (self-loop term; initializes out) ----------------
__global__ void k_selfloop(const float4* __restrict__ emb4, const float* __restrict__ dinv,
                           float4* __restrict__ out4, int N) {
    int i = blockIdx.x * blockDim.x + threadIdx.x;   // over N * (EMB_DIM/4)
    int node = i >> 5;                               // EMB_DIM/4 == 32 quads per row
    int q    = i & 31;
    if (node < N) {
        float s = dinv[node];
        s *= s;
        float4 v = emb4[node * 32 + q];
        v.x *= s; v.y *= s; v.z *= s; v.w *= s;
        out4[node * 32 + q] = v;
    }
}

// ---------------- K5: edge scatter via f32 WMMA ----------------
// One wave32 handles a FULL tile of 16 edges (branchless; no EXEC games around
// the WMMAs). For each 16-wide column slab t of the 128-dim embedding:
//   D(16x16) = diag(norm[0..15]) * EmbRows(16x16)
// as a chain of four V_WMMA_F32_16X16X4_F32 ops (full f32 precision), then
// scatter-add D into out[col[m]][t*16 + n] with f32 global atomics.
//
// f32 A-matrix 16x4 layout (ISA 7.12.2): VGPR0 holds K=0 (lanes 0-15) / K=2
// (lanes 16-31); VGPR1 holds K=1 / K=3. B 4x16 mirrors that striping with
// columns across lanes. D 16x16: VGPR v holds M = v + 8*(lane>=16), N = lane%16.
//
// A <16-edge tail (empty when E % 16 == 0, as here) is handled by wave 0 with
// a simple wave-per-edge float4 path.
__global__ void k_scatter_wmma(const int* __restrict__ row, const int* __restrict__ col,
                               const float* __restrict__ emb, const float* __restrict__ dinv,
                               float* __restrict__ out, int E) {
    const int lane = threadIdx.x & 31;
    const int half = lane >> 4;          // 0: lanes 0-15, 1: lanes 16-31
    const int lm   = lane & 15;
    const int wave   = (blockIdx.x * blockDim.x + threadIdx.x) >> 5;
    const int nwaves = (gridDim.x * blockDim.x) >> 5;
    const int ntiles = E >> 4;           // full tiles only

    // Lane-constant diagonal selection masks for the A slices.
    bool selx[4], sely[4];
#pragma unroll
    for (int kb = 0; kb < 4; ++kb) {
        int k0 = kb * 4 + 2 * half;
        selx[kb] = (lm == k0);
        sely[kb] = (lm == k0 + 1);
    }

    for (int tile = wave; tile < ntiles; tile += nwaves) {
        // Both lane-halves mirror the same 16 edges of this tile.
        int e     = tile * 16 + lm;
        int r     = row[e];
        int c     = col[e];
        float nrm = dinv[r] * dinv[c];

        // A slices: A_kb[m][k] = (m == kb*4+k) ? norm[m] : 0
        v2f a[4];
#pragma unroll
        for (int kb = 0; kb < 4; ++kb) {
            a[kb].x = selx[kb] ? nrm : 0.0f;
            a[kb].y = sely[kb] ? nrm : 0.0f;
        }

        // t-invariant element offsets: B gather rows and D scatter targets.
        int boff0[4], boff1[4];
#pragma unroll
        for (int kb = 0; kb < 4; ++kb) {
            int j0    = kb * 4 + 2 * half;
            boff0[kb] = __shfl(r, j0, 32) * EMB_DIM + lm;      // K = 2*half
            boff1[kb] = __shfl(r, j0 + 1, 32) * EMB_DIM + lm;  // K = 2*half+1
        }
        int ooff[8];
#pragma unroll
        for (int v = 0; v < 8; ++v) {
            int m   = v + 8 * half;
            ooff[v] = __shfl(c, m, 32) * EMB_DIM + lm;
        }

#pragma unroll
        for (int t = 0; t < 8; ++t) {           // 8 slabs of 16 columns
            v8f acc = {};
#pragma unroll
            for (int kb = 0; kb < 4; ++kb) {
                v2f b;
                b.x = emb[boff0[kb] + t * 16];
                b.y = emb[boff1[kb] + t * 16];
                acc = __builtin_amdgcn_wmma_f32_16x16x4_f32(
                        /*neg_a=*/false, a[kb], /*neg_b=*/false, b,
                        /*c_mod=*/(short)0, acc, /*reuse_a=*/false, /*reuse_b=*/false);
            }
#pragma unroll
            for (int v = 0; v < 8; ++v) {
                atomicAdd(&out[ooff[v] + t * 16], acc[v]);
            }
        }
    }

    // Tail: remaining E % 16 edges (none when E is a multiple of 16).
    if (wave == 0) {
        for (int e = ntiles * 16; e < E; ++e) {
            int r = row[e];
            int c = col[e];
            float nrm = dinv[r] * dinv[c];
            float4 v4 = ((const float4*)(emb + r * EMB_DIM))[lane];
            float* op = out + c * EMB_DIM + lane * 4;
            atomicAdd(op + 0, nrm * v4.x);
            atomicAdd(op + 1, nrm * v4.y);
            atomicAdd(op + 2, nrm * v4.z);
            atomicAdd(op + 3, nrm * v4.w);
        }
    }
}

extern "C" void kernel_launch(void* const* d_in, const int* in_sizes, int n_in,
                              void* d_out, int out_size, void* d_ws, size_t ws_size,
                              hipStream_t stream) {
    const int*   edges = (const int*)d_in[0];     // [2, E] int32
    const float* emb   = (const float*)d_in[1];   // [N, 128] f32
    const int E = in_sizes[0] / 2;
    const int N = in_sizes[1] / EMB_DIM;
    const int* row = edges;
    const int* col = edges + E;
    float* out = (float*)d_out;

    // Workspace: deg (N u32) | dinv (N f32)
    unsigned* deg  = (unsigned*)d_ws;
    float*    dinv = (float*)(deg + N);

    k_deg_init<<<(N + 255) / 256, 256, 0, stream>>>(deg, N);
    k_deg_count<<<(E + 255) / 256, 256, 0, stream>>>(row, deg, E);
    k_dinv<<<(N + 255) / 256, 256, 0, stream>>>(deg, dinv, N);

    int selfThreads = N * (EMB_DIM / 4);
    k_selfloop<<<(selfThreads + 255) / 256, 256, 0, stream>>>(
        (const float4*)emb, dinv, (float4*)out, N);

    int ntiles = E / 16;                  // one wave32 per 16-edge tile
    int blocks = (ntiles + 7) / 8;        // 256 threads = 8 waves per block
    if (blocks < 1) blocks = 1;
    k_scatter_wmma<<<blocks, 256, 0, stream>>>(row, col, emb, dinv, out, E);
}